// EquivariantMGNConv_78915729097030
// MI455X (gfx1250) — compile-verified
//
#include <hip/hip_runtime.h>
#include <math.h>

#define S 8
#define V 4
#define D 20          // S + 3*V
#define OS 12         // S + V outputs of tp_gate scalar path
#define NN 25000
#define NE 400000
#define EPSV 1e-5f

// edge tp_gate feature dims
#define K1S 168       // 17*8 outer + 8*4 dot
#define K1V 164       // 17*4 + 8*8 + 8*4 cross
// node tp_gate feature dims
#define K2S 80        // 8*8 + 4*4 dot
#define K2V 80        // 8*4 + 4*8 + 4*4 cross

typedef __attribute__((ext_vector_type(2))) float v2f;
typedef __attribute__((ext_vector_type(8))) float v8f;

__device__ __forceinline__ float sigmoidf_(float x) { return 1.0f / (1.0f + expf(-x)); }

// ---------------- edge tp_gate feature helpers ----------------
// All region boundaries are multiples of 4, so with kc compile-time these
// branches fold to constants regardless of the runtime half-index h.
__device__ __forceinline__ float feat1s(const float* S1e, const float* S2e, const float* DOTe,
                                        int f) {
  if (f < 136) return S1e[f >> 3] * S2e[f & 7];
  return DOTe[f - 136];
}

template <int C>
__device__ __forceinline__ float feat1v(const float* S1e, const float* S2e, const float* V1e,
                                        const float* V2e, const float* CRSe, int f) {
  if (f < 68) return S1e[f >> 2] * V2e[(f & 3) * 3 + C];
  if (f < 132) { int g = f - 68; return V1e[(g >> 3) * 3 + C] * S2e[g & 7]; }
  return CRSe[(f - 132) * 3 + C];
}

template <int C>
__device__ __forceinline__ v8f edge_vpath(const float* S1e, const float* S2e, const float* V1e,
                                          const float* V2e, const float* CRSe,
                                          const float* WVp, int h, int n) {
  v8f acc = {};
  const float* wb = WVp + ((size_t)h * 16 + n) * 2;  // packed [kc][h][n][2]
#pragma unroll
  for (int kc = 0; kc < K1V / 4; ++kc) {
    int f0 = kc * 4 + 2 * h;
    v2f a;
    a.x = feat1v<C>(S1e, S2e, V1e, V2e, CRSe, f0);
    a.y = feat1v<C>(S1e, S2e, V1e, V2e, CRSe, f0 + 1);
    v2f b = *(const v2f*)(wb + kc * 64);
    acc = __builtin_amdgcn_wmma_f32_16x16x4_f32(false, a, false, b, (short)0, acc, false, false);
  }
  return acc;
}

// ---------------- node tp_gate feature helpers ----------------
__device__ __forceinline__ float feat2s(const float* S1e, const float* S2e, const float* DOTe,
                                        int f) {
  if (f < 64) return S1e[f >> 3] * S2e[f & 7];
  return DOTe[f - 64];
}

template <int C>
__device__ __forceinline__ float feat2v(const float* S1e, const float* S2e, const float* V1e,
                                        const float* V2e, const float* CRSe, int f) {
  if (f < 32) return S1e[f >> 2] * V2e[(f & 3) * 3 + C];
  if (f < 64) { int g = f - 32; return V1e[(g >> 3) * 3 + C] * S2e[g & 7]; }
  return CRSe[(f - 64) * 3 + C];
}

template <int C>
__device__ __forceinline__ v8f node_vpath(const float* S1e, const float* S2e, const float* V1e,
                                          const float* V2e, const float* CRSe,
                                          const float* WVp, int h, int n) {
  v8f acc = {};
  const float* wb = WVp + ((size_t)h * 16 + n) * 2;
#pragma unroll
  for (int kc = 0; kc < K2V / 4; ++kc) {
    int f0 = kc * 4 + 2 * h;
    v2f a;
    a.x = feat2v<C>(S1e, S2e, V1e, V2e, CRSe, f0);
    a.y = feat2v<C>(S1e, S2e, V1e, V2e, CRSe, f0 + 1);
    v2f b = *(const v2f*)(wb + kc * 64);
    acc = __builtin_amdgcn_wmma_f32_16x16x4_f32(false, a, false, b, (short)0, acc, false, false);
  }
  return acc;
}

// ================= Kernel 1: edge tp_gate + stats =================
// 64 threads = 2 waves; each wave handles 16 edges.
__global__ __launch_bounds__(64) void k_edge_tp(
    const float* __restrict__ x, const float* __restrict__ ea, const float* __restrict__ elen,
    const int* __restrict__ eidx,
    const float* __restrict__ Wss, const float* __restrict__ Wsv, const float* __restrict__ Wvs,
    const float* __restrict__ Wvvs, const float* __restrict__ Wvvv,
    float* __restrict__ msmv, float* __restrict__ estat) {
  __shared__ float WSs[K1S * 16];  // packed [kc][h][n][2]
  __shared__ float WVs[K1V * 16];  // packed [kc][h][n][2]
  __shared__ float ST[2][3024];
  __shared__ float BST[20];

  const int tid = threadIdx.x;
  // stage weights in packed layout: idx = ((kc*2 + h)*16 + n)*2 + j ; f = 4kc + 2h + j
  for (int idx = tid; idx < K1S * 16; idx += 64) {
    int j = idx & 1, n = (idx >> 1) & 15, hh = (idx >> 5) & 1, kc = idx >> 6;
    int f = kc * 4 + 2 * hh + j;
    float w = 0.0f;
    if (n < OS) w = (f < 136) ? Wss[f * OS + n] : Wvvs[(f - 136) * OS + n];
    WSs[idx] = w;
  }
  for (int idx = tid; idx < K1V * 16; idx += 64) {
    int j = idx & 1, n = (idx >> 1) & 15, hh = (idx >> 5) & 1, kc = idx >> 6;
    int f = kc * 4 + 2 * hh + j;
    float w = 0.0f;
    if (n < V) {
      if (f < 68) w = Wsv[f * V + n];
      else if (f < 132) w = Wvs[(f - 68) * V + n];
      else w = Wvvv[(f - 132) * V + n];
    }
    WVs[idx] = w;
  }
  if (tid < 20) BST[tid] = 0.0f;

  const int wave = tid >> 5;
  const int lane = tid & 31;
  const int e = lane & 15;
  const int h = lane >> 4;
  const int n = e;  // WMMA column for B / C / D
  const int tileBase = (blockIdx.x * 2 + wave) * 16;
  const int eg = tileBase + e;

  float* S1 = &ST[wave][0];     // [16][17]
  float* S2 = &ST[wave][272];   // [16][8]
  float* V1 = &ST[wave][400];   // [16][24]
  float* V2 = &ST[wave][784];   // [16][12]
  float* DOT = &ST[wave][976];  // [16][32]
  float* CRS = &ST[wave][1488]; // [16][96]

  if (h == 0) {
    int i = eidx[eg], j = eidx[NE + eg];
#pragma unroll
    for (int u = 0; u < 8; ++u) { S1[e * 17 + u] = x[i * D + u]; S1[e * 17 + 8 + u] = x[j * D + u]; }
    S1[e * 17 + 16] = elen[eg];
#pragma unroll
    for (int t = 0; t < 12; ++t) { V1[e * 24 + t] = x[i * D + 8 + t]; V1[e * 24 + 12 + t] = x[j * D + 8 + t]; }
  } else {
#pragma unroll
    for (int u = 0; u < 8; ++u) S2[e * 8 + u] = ea[eg * D + u];
#pragma unroll
    for (int t = 0; t < 12; ++t) V2[e * 12 + t] = ea[eg * D + 8 + t];
  }
  __syncthreads();
  if (h == 0) {
#pragma unroll
    for (int u = 0; u < 8; ++u)
#pragma unroll
      for (int v = 0; v < 4; ++v) {
        float d = 0.f;
        for (int c = 0; c < 3; ++c) d += V1[e * 24 + u * 3 + c] * V2[e * 12 + v * 3 + c];
        DOT[e * 32 + u * 4 + v] = d;
      }
  } else {
#pragma unroll
    for (int u = 0; u < 8; ++u)
#pragma unroll
      for (int v = 0; v < 4; ++v) {
        float a0 = V1[e * 24 + u * 3], a1 = V1[e * 24 + u * 3 + 1], a2 = V1[e * 24 + u * 3 + 2];
        float b0 = V2[e * 12 + v * 3], b1 = V2[e * 12 + v * 3 + 1], b2 = V2[e * 12 + v * 3 + 2];
        int base = e * 96 + (u * 4 + v) * 3;
        CRS[base + 0] = a1 * b2 - a2 * b1;
        CRS[base + 1] = a2 * b0 - a0 * b2;
        CRS[base + 2] = a0 * b1 - a1 * b0;
      }
  }
  __syncthreads();

  const float* S1e = S1 + e * 17;
  const float* S2e = S2 + e * 8;
  const float* V1e = V1 + e * 24;
  const float* V2e = V2 + e * 12;
  const float* DOTe = DOT + e * 32;
  const float* CRSe = CRS + e * 96;

  // scalar path: [16 edges] x [K1S] x [16 cols]
  v8f accS = {};
  {
    const float* wb = WSs + ((size_t)h * 16 + n) * 2;
#pragma unroll
    for (int kc = 0; kc < K1S / 4; ++kc) {
      int f0 = kc * 4 + 2 * h;
      v2f a;
      a.x = feat1s(S1e, S2e, DOTe, f0);
      a.y = feat1s(S1e, S2e, DOTe, f0 + 1);
      v2f b = *(const v2f*)(wb + kc * 64);
      accS = __builtin_amdgcn_wmma_f32_16x16x4_f32(false, a, false, b, (short)0, accS, false, false);
    }
  }
  // vector path: one GEMM per spatial component (shared weights)
  v8f accV0 = edge_vpath<0>(S1e, S2e, V1e, V2e, CRSe, WVs, h, n);
  v8f accV1 = edge_vpath<1>(S1e, S2e, V1e, V2e, CRSe, WVs, h, n);
  v8f accV2 = edge_vpath<2>(S1e, S2e, V1e, V2e, CRSe, WVs, h, n);

  // epilogue: SiLU on channels 0..7, sigmoid gates on 8..11 applied to v-path
  float sumS = 0.f, sumS2 = 0.f, sumV = 0.f;
#pragma unroll
  for (int r = 0; r < 8; ++r) {
    int em = tileBase + r + 8 * h;
    float sv = accS[r];
    float sg = sigmoidf_(sv);
    float gate = __shfl(sg, (lane & 16) | (8 + (n & 3)), 32);
    if (n < 8) {
      float silu = sv * sg;
      msmv[em * D + n] = silu;
      sumS += silu;
      sumS2 += silu * silu;
    }
    if (n < 4) {
      float v0 = accV0[r] * gate, v1 = accV1[r] * gate, v2 = accV2[r] * gate;
      msmv[em * D + 8 + n * 3 + 0] = v0;
      msmv[em * D + 8 + n * 3 + 1] = v1;
      msmv[em * D + 8 + n * 3 + 2] = v2;
      sumV += v0 * v0 + v1 * v1 + v2 * v2;
    }
  }
  sumS += __shfl_xor(sumS, 16, 32);
  sumS2 += __shfl_xor(sumS2, 16, 32);
  sumV += __shfl_xor(sumV, 16, 32);
  if (h == 0) {
    if (n < 8) { atomicAdd(&BST[n], sumS); atomicAdd(&BST[8 + n], sumS2); }
    if (n < 4) atomicAdd(&BST[16 + n], sumV);
  }
  __syncthreads();
  if (tid < 20) atomicAdd(&estat[tid], BST[tid]);
}

// ================= Kernel 2/5: finalize batchnorm params =================
__global__ void k_finalize_bn(const float* __restrict__ stat, const float* __restrict__ wsc,
                              const float* __restrict__ bsc, const float* __restrict__ wvc,
                              float* __restrict__ bn, float count) {
  int t = threadIdx.x;
  if (t < 8) {
    float mu = stat[t] / count;
    float var = stat[8 + t] / count - mu * mu;
    float is = rsqrtf(var + EPSV);
    float sc = wsc[t] * is;
    bn[t] = sc;
    bn[8 + t] = bsc[t] - mu * sc;
  } else if (t < 12) {
    int u = t - 8;
    bn[16 + u] = wvc[u] * rsqrtf(stat[16 + u] / (3.0f * count) + EPSV);
  }
}

// ================= Kernel 3: edge BN + residual + node tp_gate + segment-sum =================
// 128 threads = 4 waves; each wave handles 16 edges.
__global__ __launch_bounds__(128) void k_node_tp(
    const float* __restrict__ x, const float* __restrict__ ea, const int* __restrict__ eidx,
    const float* __restrict__ msmv, const float* __restrict__ ebn,
    const float* __restrict__ Wss, const float* __restrict__ Wsv, const float* __restrict__ Wvs,
    const float* __restrict__ Wvvs, const float* __restrict__ Wvvv,
    float* __restrict__ agg, float* __restrict__ eout) {
  __shared__ float WSs[K2S * 16];  // packed [kc][h][n][2]
  __shared__ float WVs[K2V * 16];  // packed [kc][h][n][2]
  __shared__ float ST[4][1664];
  __shared__ int NI[4][16];

  const int tid = threadIdx.x;
  for (int idx = tid; idx < K2S * 16; idx += 128) {
    int j = idx & 1, n = (idx >> 1) & 15, hh = (idx >> 5) & 1, kc = idx >> 6;
    int f = kc * 4 + 2 * hh + j;
    float w = 0.0f;
    if (n < OS) w = (f < 64) ? Wss[f * OS + n] : Wvvs[(f - 64) * OS + n];
    WSs[idx] = w;
  }
  for (int idx = tid; idx < K2V * 16; idx += 128) {
    int j = idx & 1, n = (idx >> 1) & 15, hh = (idx >> 5) & 1, kc = idx >> 6;
    int f = kc * 4 + 2 * hh + j;
    float w = 0.0f;
    if (n < V) {
      if (f < 32) w = Wsv[f * V + n];
      else if (f < 64) w = Wvs[(f - 32) * V + n];
      else w = Wvvv[(f - 64) * V + n];
    }
    WVs[idx] = w;
  }

  const int wave = tid >> 5;
  const int lane = tid & 31;
  const int e = lane & 15;
  const int h = lane >> 4;
  const int n = e;
  const int tileBase = (blockIdx.x * 4 + wave) * 16;
  const int eg = tileBase + e;

  float* S1 = &ST[wave][0];    // [16][8]  xs[i]
  float* S2 = &ST[wave][128];  // [16][8]  es2
  float* V1 = &ST[wave][256];  // [16][12] xv[i]
  float* V2 = &ST[wave][448];  // [16][12] ev2
  float* DOT = &ST[wave][640]; // [16][16]
  float* CRS = &ST[wave][896]; // [16][48]

  if (h == 0) {
    int i = eidx[eg];
    NI[wave][e] = i;
#pragma unroll
    for (int u = 0; u < 8; ++u) S1[e * 8 + u] = x[i * D + u];
#pragma unroll
    for (int t = 0; t < 12; ++t) V1[e * 12 + t] = x[i * D + 8 + t];
  } else {
    // edge BN affine + residual; also writes edge_out
#pragma unroll
    for (int u = 0; u < 8; ++u) {
      float v = msmv[eg * D + u] * ebn[u] + ebn[8 + u] + ea[eg * D + u];
      S2[e * 8 + u] = v;
      eout[eg * D + u] = v;
    }
#pragma unroll
    for (int t = 0; t < 12; ++t) {
      float v = msmv[eg * D + 8 + t] * ebn[16 + t / 3] + ea[eg * D + 8 + t];
      V2[e * 12 + t] = v;
      eout[eg * D + 8 + t] = v;
    }
  }
  __syncthreads();
  if (h == 0) {
#pragma unroll
    for (int u = 0; u < 4; ++u)
#pragma unroll
      for (int v = 0; v < 4; ++v) {
        float d = 0.f;
        for (int c = 0; c < 3; ++c) d += V1[e * 12 + u * 3 + c] * V2[e * 12 + v * 3 + c];
        DOT[e * 16 + u * 4 + v] = d;
      }
  } else {
#pragma unroll
    for (int u = 0; u < 4; ++u)
#pragma unroll
      for (int v = 0; v < 4; ++v) {
        float a0 = V1[e * 12 + u * 3], a1 = V1[e * 12 + u * 3 + 1], a2 = V1[e * 12 + u * 3 + 2];
        float b0 = V2[e * 12 + v * 3], b1 = V2[e * 12 + v * 3 + 1], b2 = V2[e * 12 + v * 3 + 2];
        int base = e * 48 + (u * 4 + v) * 3;
        CRS[base + 0] = a1 * b2 - a2 * b1;
        CRS[base + 1] = a2 * b0 - a0 * b2;
        CRS[base + 2] = a0 * b1 - a1 * b0;
      }
  }
  __syncthreads();

  const float* S1e = S1 + e * 8;
  const float* S2e = S2 + e * 8;
  const float* V1e = V1 + e * 12;
  const float* V2e = V2 + e * 12;
  const float* DOTe = DOT + e * 16;
  const float* CRSe = CRS + e * 48;

  v8f accS = {};
  {
    const float* wb = WSs + ((size_t)h * 16 + n) * 2;
#pragma unroll
    for (int kc = 0; kc < K2S / 4; ++kc) {
      int f0 = kc * 4 + 2 * h;
      v2f a;
      a.x = feat2s(S1e, S2e, DOTe, f0);
      a.y = feat2s(S1e, S2e, DOTe, f0 + 1);
      v2f b = *(const v2f*)(wb + kc * 64);
      accS = __builtin_amdgcn_wmma_f32_16x16x4_f32(false, a, false, b, (short)0, accS, false, false);
    }
  }
  v8f accV0 = node_vpath<0>(S1e, S2e, V1e, V2e, CRSe, WVs, h, n);
  v8f accV1 = node_vpath<1>(S1e, S2e, V1e, V2e, CRSe, WVs, h, n);
  v8f accV2 = node_vpath<2>(S1e, S2e, V1e, V2e, CRSe, WVs, h, n);

#pragma unroll
  for (int r = 0; r < 8; ++r) {
    int m = r + 8 * h;
    int node = NI[wave][m];
    float sv = accS[r];
    float sg = sigmoidf_(sv);
    float gate = __shfl(sg, (lane & 16) | (8 + (n & 3)), 32);
    if (n < 8) atomicAdd(&agg[node * D + n], sv * sg);
    if (n < 4) {
      atomicAdd(&agg[node * D + 8 + n * 3 + 0], accV0[r] * gate);
      atomicAdd(&agg[node * D + 8 + n * 3 + 1], accV1[r] * gate);
      atomicAdd(&agg[node * D + 8 + n * 3 + 2], accV2[r] * gate);
    }
  }
}

// ================= Kernel 4: node BN statistics =================
__global__ __launch_bounds__(256) void k_node_stats(const float* __restrict__ agg,
                                                    float* __restrict__ nstat) {
  __shared__ float BST[20];
  int tid = threadIdx.x;
  if (tid < 20) BST[tid] = 0.f;
  __syncthreads();
  int node = blockIdx.x * 256 + tid;
  float ss[8] = {}, s2[8] = {}, vv[4] = {};
  if (node < NN) {
#pragma unroll
    for (int k = 0; k < 8; ++k) { float t = agg[node * D + k]; ss[k] = t; s2[k] = t * t; }
#pragma unroll
    for (int t = 0; t < 12; ++t) { float z = agg[node * D + 8 + t]; vv[t / 3] += z * z; }
  }
#pragma unroll
  for (int off = 16; off >= 1; off >>= 1) {
#pragma unroll
    for (int k = 0; k < 8; ++k) { ss[k] += __shfl_xor(ss[k], off, 32); s2[k] += __shfl_xor(s2[k], off, 32); }
#pragma unroll
    for (int u = 0; u < 4; ++u) vv[u] += __shfl_xor(vv[u], off, 32);
  }
  if ((tid & 31) == 0) {
#pragma unroll
    for (int k = 0; k < 8; ++k) { atomicAdd(&BST[k], ss[k]); atomicAdd(&BST[8 + k], s2[k]); }
#pragma unroll
    for (int u = 0; u < 4; ++u) atomicAdd(&BST[16 + u], vv[u]);
  }
  __syncthreads();
  if (tid < 20) atomicAdd(&nstat[tid], BST[tid]);
}

// ================= Kernel 6: node BN + skip -> node_attr =================
__global__ __launch_bounds__(256) void k_node_out(const float* __restrict__ agg,
                                                  const float* __restrict__ x,
                                                  const float* __restrict__ nbn,
                                                  float* __restrict__ out) {
  int idx = blockIdx.x * 256 + threadIdx.x;
  if (idx >= NN * D) return;
  int k = idx % D;
  float a = agg[idx];
  float r = (k < 8) ? (a * nbn[k] + nbn[8 + k]) : (a * nbn[16 + (k - 8) / 3]);
  out[idx] = r + x[idx];
}

extern "C" void kernel_launch(void* const* d_in, const int* in_sizes, int n_in,
                              void* d_out, int out_size, void* d_ws, size_t ws_size,
                              hipStream_t stream) {
  (void)in_sizes; (void)n_in; (void)out_size; (void)ws_size;
  const float* x = (const float*)d_in[0];
  const float* ea = (const float*)d_in[1];
  const float* elen = (const float*)d_in[2];
  const float* We_ss = (const float*)d_in[3];
  const float* We_sv = (const float*)d_in[4];
  const float* We_vs = (const float*)d_in[5];
  const float* We_vvs = (const float*)d_in[6];
  const float* We_vvv = (const float*)d_in[7];
  const float* bne_ws = (const float*)d_in[8];
  const float* bne_bs = (const float*)d_in[9];
  const float* bne_wv = (const float*)d_in[10];
  const float* Wn_ss = (const float*)d_in[11];
  const float* Wn_sv = (const float*)d_in[12];
  const float* Wn_vs = (const float*)d_in[13];
  const float* Wn_vvs = (const float*)d_in[14];
  const float* Wn_vvv = (const float*)d_in[15];
  const float* bnn_ws = (const float*)d_in[16];
  const float* bnn_bs = (const float*)d_in[17];
  const float* bnn_wv = (const float*)d_in[18];
  const int* eidx = (const int*)d_in[19];

  float* w = (float*)d_ws;
  float* msmv = w;                               // E*20
  float* agg = w + (size_t)NE * D;               // N*20
  float* estat = agg + (size_t)NN * D;           // 20
  float* nstat = estat + 20;                     // 20
  float* ebn = nstat + 20;                       // 20
  float* nbn = ebn + 20;                         // 20
  float* outf = (float*)d_out;
  float* eout = outf + (size_t)NN * D;

  // zero agg + both stat accumulators (contiguous)
  hipMemsetAsync(agg, 0, ((size_t)NN * D + 40) * sizeof(float), stream);

  k_edge_tp<<<NE / 32, 64, 0, stream>>>(x, ea, elen, eidx, We_ss, We_sv, We_vs, We_vvs, We_vvv,
                                        msmv, estat);
  k_finalize_bn<<<1, 32, 0, stream>>>(estat, bne_ws, bne_bs, bne_wv, ebn, (float)NE);
  k_node_tp<<<NE / 64, 128, 0, stream>>>(x, ea, eidx, msmv, ebn, Wn_ss, Wn_sv, Wn_vs, Wn_vvs,
                                         Wn_vvv, agg, eout);
  k_node_stats<<<(NN + 255) / 256, 256, 0, stream>>>(agg, nstat);
  k_finalize_bn<<<1, 32, 0, stream>>>(nstat, bnn_ws, bnn_bs, bnn_wv, nbn, (float)NN);
  k_node_out<<<(NN * D + 255) / 256, 256, 0, stream>>>(agg, x, nbn, outf);
}